// S5SSM_35545149341915
// MI455X (gfx1250) — compile-verified
//
#include <hip/hip_runtime.h>
#include <hip/hip_bf16.h>

// ---------------------------------------------------------------------------
// S5 SSM (L=8192, H=512, P=256) for MI455X / gfx1250 (wave32, WMMA).
// Pipeline: prep -> GEMM1 (Bu = u @ Bbar^T, f32 WMMA, 4xM blocking, SW-pipelined)
//        -> 2-level chunked reset-scan
//        -> GEMM2 (out = 2*Re(x @ C^T) + u*D, f32 WMMA, 4xM x 2xN, SW-pipelined).
// All heavy math uses V_WMMA_F32_16X16X4_F32 to stay in fp32.
// ---------------------------------------------------------------------------

#define L_SEQ 8192
#define H_DIM 512
#define P_DIM 256
#define CHUNK 64
#define NCHUNK (L_SEQ / CHUNK)   // 128

typedef __attribute__((ext_vector_type(2))) float v2f;
typedef __attribute__((ext_vector_type(8))) float v8f;

#define WMMA_F32(A, B, Cacc) \
    __builtin_amdgcn_wmma_f32_16x16x4_f32(false, (A), false, (B), (short)0, (Cacc), false, false)

// ---------------------------------------------------------------------------
// Prep A: Lambda_bar, B_bar (stored [P][H] re/im planes)
// ---------------------------------------------------------------------------
__global__ __launch_bounds__(256) void s5_prep_b(
    const float* __restrict__ Lam_re, const float* __restrict__ Lam_im,
    const float* __restrict__ B,      const float* __restrict__ log_step,
    float* __restrict__ Bbar_re, float* __restrict__ Bbar_im,
    float* __restrict__ LamB_re, float* __restrict__ LamB_im)
{
    int idx = blockIdx.x * blockDim.x + threadIdx.x;   // P*H threads
    if (idx >= P_DIM * H_DIM) return;
    int p = idx >> 9;          // / H_DIM
    int h = idx & (H_DIM - 1);

    float dt  = __expf(log_step[p]);
    float lre = Lam_re[p], lim = Lam_im[p];
    float mag = __expf(lre * dt);
    float lbr = mag * __cosf(lim * dt);
    float lbi = mag * __sinf(lim * dt);
    if (h == 0) { LamB_re[p] = lbr; LamB_im[p] = lbi; }

    // coef = (Lambda_bar - 1) / Lambda   (complex divide)
    float nre = lbr - 1.0f, nim = lbi;
    float inv = 1.0f / (lre * lre + lim * lim);
    float cre = (nre * lre + nim * lim) * inv;
    float cim = (nim * lre - nre * lim) * inv;

    float br = B[(p * H_DIM + h) * 2 + 0];
    float bi = B[(p * H_DIM + h) * 2 + 1];
    Bbar_re[idx] = cre * br - cim * bi;
    Bbar_im[idx] = cre * bi + cim * br;
}

// ---------------------------------------------------------------------------
// Prep B: C_tilde repack -> Cw_re[h][p] = C[h][p][0], Cw_imn[h][p] = -C[h][p][1]
// ---------------------------------------------------------------------------
__global__ __launch_bounds__(256) void s5_prep_c(
    const float* __restrict__ C, float* __restrict__ Cw_re, float* __restrict__ Cw_imn)
{
    int idx = blockIdx.x * blockDim.x + threadIdx.x;   // H*P threads
    if (idx >= H_DIM * P_DIM) return;
    Cw_re[idx]  =  C[2 * idx + 0];
    Cw_imn[idx] = -C[2 * idx + 1];
}

// ---------------------------------------------------------------------------
// GEMM1: Bu[l][p] = sum_h u[l][h] * Bbar[p][h]   (re and im)
// 4 M-tiles per wave sharing B_re/B_im frags; software-pipelined (double-
// buffered fragments) so loads of step k+1 overlap the 8 WMMAs of step k.
// A frag (16x4 f32): lane(0..15)=M, VGPR0/1 = K{0,1}; lanes16..31 = K{2,3}.
// B frag (4x16 f32): lane(0..15)=N, VGPR0/1 = K{0,1}; lanes16..31 = K{2,3}.
// ---------------------------------------------------------------------------
__global__ __launch_bounds__(256) void s5_gemm_bu(
    const float* __restrict__ u,
    const float* __restrict__ Wre, const float* __restrict__ Wim,
    float* __restrict__ BuRe, float* __restrict__ BuIm)
{
    const int lane = threadIdx.x & 31;
    const int wave = threadIdx.x >> 5;
    const int Mquad = (blockIdx.y * 8 + wave) * 64;   // 4 M-tiles over L
    const int Nbase = blockIdx.x * 16;                // over P
    const int mr   = lane & 15;
    const int kofs = (lane >> 4) * 2;

    const float* ua0 = u + (size_t)(Mquad +  0 + mr) * H_DIM + kofs;
    const float* ua1 = u + (size_t)(Mquad + 16 + mr) * H_DIM + kofs;
    const float* ua2 = u + (size_t)(Mquad + 32 + mr) * H_DIM + kofs;
    const float* ua3 = u + (size_t)(Mquad + 48 + mr) * H_DIM + kofs;
    const float* wr  = Wre + (size_t)(Nbase + mr) * H_DIM + kofs;
    const float* wi  = Wim + (size_t)(Nbase + mr) * H_DIM + kofs;

    v8f accR[4] = {}; v8f accI[4] = {};

    // prologue: fragments for k = 0
    v2f br = *(const v2f*)(wr);
    v2f bi = *(const v2f*)(wi);
    v2f a0 = *(const v2f*)(ua0);
    v2f a1 = *(const v2f*)(ua1);
    v2f a2 = *(const v2f*)(ua2);
    v2f a3 = *(const v2f*)(ua3);

    for (int k = 0; k < H_DIM - 4; k += 4) {
        // prefetch k+4 before consuming k
        v2f brn = *(const v2f*)(wr + k + 4);
        v2f bin = *(const v2f*)(wi + k + 4);
        v2f a0n = *(const v2f*)(ua0 + k + 4);
        v2f a1n = *(const v2f*)(ua1 + k + 4);
        v2f a2n = *(const v2f*)(ua2 + k + 4);
        v2f a3n = *(const v2f*)(ua3 + k + 4);

        accR[0] = WMMA_F32(a0, br, accR[0]);
        accI[0] = WMMA_F32(a0, bi, accI[0]);
        accR[1] = WMMA_F32(a1, br, accR[1]);
        accI[1] = WMMA_F32(a1, bi, accI[1]);
        accR[2] = WMMA_F32(a2, br, accR[2]);
        accI[2] = WMMA_F32(a2, bi, accI[2]);
        accR[3] = WMMA_F32(a3, br, accR[3]);
        accI[3] = WMMA_F32(a3, bi, accI[3]);

        br = brn; bi = bin;
        a0 = a0n; a1 = a1n; a2 = a2n; a3 = a3n;
    }
    // epilogue: last K step
    accR[0] = WMMA_F32(a0, br, accR[0]);
    accI[0] = WMMA_F32(a0, bi, accI[0]);
    accR[1] = WMMA_F32(a1, br, accR[1]);
    accI[1] = WMMA_F32(a1, bi, accI[1]);
    accR[2] = WMMA_F32(a2, br, accR[2]);
    accI[2] = WMMA_F32(a2, bi, accI[2]);
    accR[3] = WMMA_F32(a3, br, accR[3]);
    accI[3] = WMMA_F32(a3, bi, accI[3]);

    // C/D layout: VGPR i, lanes<16 -> M=i ; lanes>=16 -> M=i+8 ; N = lane&15
    const int col = Nbase + mr;
#pragma unroll
    for (int j = 0; j < 4; ++j) {
        const int rbase = Mquad + j * 16 + 8 * (lane >> 4);
#pragma unroll
        for (int i = 0; i < 8; ++i) {
            BuRe[(size_t)(rbase + i) * P_DIM + col] = accR[j][i];
            BuIm[(size_t)(rbase + i) * P_DIM + col] = accI[j][i];
        }
    }
}

// ---------------------------------------------------------------------------
// Scan S1: per (p, chunk) aggregate (A, b, c) of the reset-monoid over CHUNK.
//   element_t = (a_p, Bu_t, r_t);  r=1 resets: A=a, b=Bu, c=1
// ---------------------------------------------------------------------------
__global__ __launch_bounds__(P_DIM) void s5_scan_agg(
    const float* __restrict__ BuRe, const float* __restrict__ BuIm,
    const float* __restrict__ LamB_re, const float* __restrict__ LamB_im,
    const int* __restrict__ resets,
    float* __restrict__ AggAre, float* __restrict__ AggAim,
    float* __restrict__ AggBre, float* __restrict__ AggBim,
    float* __restrict__ AggC)
{
    const int p = threadIdx.x;
    const int ch = blockIdx.x;
    const float are = LamB_re[p], aim = LamB_im[p];

    float Are = 1.0f, Aim = 0.0f;    // identity
    float bre = 0.0f, bim = 0.0f;
    float c   = 0.0f;

    const int t0 = ch * CHUNK;
    for (int t = t0; t < t0 + CHUNK; ++t) {
        const float ure = BuRe[(size_t)t * P_DIM + p];
        const float uim = BuIm[(size_t)t * P_DIM + p];
        if (resets[t]) {
            Are = are; Aim = aim;
            bre = ure; bim = uim;
            c = 1.0f;
        } else {
            float nAre = are * Are - aim * Aim;
            float nAim = are * Aim + aim * Are;
            Are = nAre; Aim = nAim;
            float nbre = are * bre - aim * bim + ure;
            float nbim = are * bim + aim * bre + uim;
            bre = nbre; bim = nbim;
        }
    }
    const int o = ch * P_DIM + p;
    AggAre[o] = Are; AggAim[o] = Aim;
    AggBre[o] = bre; AggBim[o] = bim;
    AggC[o]   = c;
}

// ---------------------------------------------------------------------------
// Scan S2: serial scan across NCHUNK aggregates per p; emit incoming state of
// each chunk (exclusive prefix), seeded with hidden[p].
// ---------------------------------------------------------------------------
__global__ __launch_bounds__(P_DIM) void s5_scan_spine(
    const float* __restrict__ hidden,
    const float* __restrict__ AggAre, const float* __restrict__ AggAim,
    const float* __restrict__ AggBre, const float* __restrict__ AggBim,
    const float* __restrict__ AggC,
    float* __restrict__ XinRe, float* __restrict__ XinIm)
{
    const int p = threadIdx.x;
    float xre = hidden[p], xim = 0.0f;
    for (int k = 0; k < NCHUNK; ++k) {
        const int o = k * P_DIM + p;
        XinRe[o] = xre; XinIm[o] = xim;
        if (AggC[o] != 0.0f) {
            xre = AggBre[o]; xim = AggBim[o];
        } else {
            float Are = AggAre[o], Aim = AggAim[o];
            float nre = Are * xre - Aim * xim + AggBre[o];
            float nim = Are * xim + Aim * xre + AggBim[o];
            xre = nre; xim = nim;
        }
    }
}

// ---------------------------------------------------------------------------
// Scan S3: replay each chunk from its incoming state, write x_t to [L][P].
// ---------------------------------------------------------------------------
__global__ __launch_bounds__(P_DIM) void s5_scan_apply(
    const float* __restrict__ BuRe, const float* __restrict__ BuIm,
    const float* __restrict__ LamB_re, const float* __restrict__ LamB_im,
    const int* __restrict__ resets,
    const float* __restrict__ XinRe, const float* __restrict__ XinIm,
    float* __restrict__ Xre, float* __restrict__ Xim)
{
    const int p = threadIdx.x;
    const int ch = blockIdx.x;
    const float are = LamB_re[p], aim = LamB_im[p];

    float xre = XinRe[ch * P_DIM + p];
    float xim = XinIm[ch * P_DIM + p];

    const int t0 = ch * CHUNK;
    for (int t = t0; t < t0 + CHUNK; ++t) {
        const size_t o = (size_t)t * P_DIM + p;
        const float ure = BuRe[o], uim = BuIm[o];
        if (resets[t]) {
            xre = ure; xim = uim;
        } else {
            float nre = are * xre - aim * xim + ure;
            float nim = are * xim + aim * xre + uim;
            xre = nre; xim = nim;
        }
        Xre[o] = xre; Xim[o] = xim;
    }
}

// ---------------------------------------------------------------------------
// GEMM2 + epilogue: out[l][h] = 2*(Xre@Cw_re^T + Xim@Cw_imn^T) + u[l][h]*D[h]
// K = P = 256. 4 M-tiles x 2 N-tiles per wave, software-pipelined fragments.
// WMMAs ordered: all 8 "re" products then all 8 "im" products so that the two
// updates of each accumulator are 8 matrix ops apart (no D->C back-to-back).
// ---------------------------------------------------------------------------
__global__ __launch_bounds__(256) void s5_gemm_out(
    const float* __restrict__ Xre, const float* __restrict__ Xim,
    const float* __restrict__ Cw_re, const float* __restrict__ Cw_imn,
    const float* __restrict__ u, const float* __restrict__ D,
    float* __restrict__ out)
{
    const int lane = threadIdx.x & 31;
    const int wave = threadIdx.x >> 5;
    const int Mquad = (blockIdx.y * 8 + wave) * 64;   // 4 M-tiles over L
    const int Nbase = blockIdx.x * 32;                // 2 N-tiles over H
    const int mr   = lane & 15;
    const int kofs = (lane >> 4) * 2;

    const float* xr0 = Xre + (size_t)(Mquad +  0 + mr) * P_DIM + kofs;
    const float* xr1 = Xre + (size_t)(Mquad + 16 + mr) * P_DIM + kofs;
    const float* xr2 = Xre + (size_t)(Mquad + 32 + mr) * P_DIM + kofs;
    const float* xr3 = Xre + (size_t)(Mquad + 48 + mr) * P_DIM + kofs;
    const float* xi0 = Xim + (size_t)(Mquad +  0 + mr) * P_DIM + kofs;
    const float* xi1 = Xim + (size_t)(Mquad + 16 + mr) * P_DIM + kofs;
    const float* xi2 = Xim + (size_t)(Mquad + 32 + mr) * P_DIM + kofs;
    const float* xi3 = Xim + (size_t)(Mquad + 48 + mr) * P_DIM + kofs;
    const float* cr0 = Cw_re  + (size_t)(Nbase +  0 + mr) * P_DIM + kofs;
    const float* cr1 = Cw_re  + (size_t)(Nbase + 16 + mr) * P_DIM + kofs;
    const float* ci0 = Cw_imn + (size_t)(Nbase +  0 + mr) * P_DIM + kofs;
    const float* ci1 = Cw_imn + (size_t)(Nbase + 16 + mr) * P_DIM + kofs;

    v8f acc[4][2] = {};

    // prologue: fragments for k = 0
    v2f br0 = *(const v2f*)(cr0);
    v2f br1 = *(const v2f*)(cr1);
    v2f bi0 = *(const v2f*)(ci0);
    v2f bi1 = *(const v2f*)(ci1);
    v2f ar0 = *(const v2f*)(xr0);
    v2f ar1 = *(const v2f*)(xr1);
    v2f ar2 = *(const v2f*)(xr2);
    v2f ar3 = *(const v2f*)(xr3);
    v2f ai0 = *(const v2f*)(xi0);
    v2f ai1 = *(const v2f*)(xi1);
    v2f ai2 = *(const v2f*)(xi2);
    v2f ai3 = *(const v2f*)(xi3);

    for (int k = 0; k < P_DIM - 4; k += 4) {
        // prefetch k+4
        v2f br0n = *(const v2f*)(cr0 + k + 4);
        v2f br1n = *(const v2f*)(cr1 + k + 4);
        v2f bi0n = *(const v2f*)(ci0 + k + 4);
        v2f bi1n = *(const v2f*)(ci1 + k + 4);
        v2f ar0n = *(const v2f*)(xr0 + k + 4);
        v2f ar1n = *(const v2f*)(xr1 + k + 4);
        v2f ar2n = *(const v2f*)(xr2 + k + 4);
        v2f ar3n = *(const v2f*)(xr3 + k + 4);
        v2f ai0n = *(const v2f*)(xi0 + k + 4);
        v2f ai1n = *(const v2f*)(xi1 + k + 4);
        v2f ai2n = *(const v2f*)(xi2 + k + 4);
        v2f ai3n = *(const v2f*)(xi3 + k + 4);

        acc[0][0] = WMMA_F32(ar0, br0, acc[0][0]);
        acc[0][1] = WMMA_F32(ar0, br1, acc[0][1]);
        acc[1][0] = WMMA_F32(ar1, br0, acc[1][0]);
        acc[1][1] = WMMA_F32(ar1, br1, acc[1][1]);
        acc[2][0] = WMMA_F32(ar2, br0, acc[2][0]);
        acc[2][1] = WMMA_F32(ar2, br1, acc[2][1]);
        acc[3][0] = WMMA_F32(ar3, br0, acc[3][0]);
        acc[3][1] = WMMA_F32(ar3, br1, acc[3][1]);
        acc[0][0] = WMMA_F32(ai0, bi0, acc[0][0]);
        acc[0][1] = WMMA_F32(ai0, bi1, acc[0][1]);
        acc[1][0] = WMMA_F32(ai1, bi0, acc[1][0]);
        acc[1][1] = WMMA_F32(ai1, bi1, acc[1][1]);
        acc[2][0] = WMMA_F32(ai2, bi0, acc[2][0]);
        acc[2][1] = WMMA_F32(ai2, bi1, acc[2][1]);
        acc[3][0] = WMMA_F32(ai3, bi0, acc[3][0]);
        acc[3][1] = WMMA_F32(ai3, bi1, acc[3][1]);

        br0 = br0n; br1 = br1n; bi0 = bi0n; bi1 = bi1n;
        ar0 = ar0n; ar1 = ar1n; ar2 = ar2n; ar3 = ar3n;
        ai0 = ai0n; ai1 = ai1n; ai2 = ai2n; ai3 = ai3n;
    }
    // epilogue: last K step
    acc[0][0] = WMMA_F32(ar0, br0, acc[0][0]);
    acc[0][1] = WMMA_F32(ar0, br1, acc[0][1]);
    acc[1][0] = WMMA_F32(ar1, br0, acc[1][0]);
    acc[1][1] = WMMA_F32(ar1, br1, acc[1][1]);
    acc[2][0] = WMMA_F32(ar2, br0, acc[2][0]);
    acc[2][1] = WMMA_F32(ar2, br1, acc[2][1]);
    acc[3][0] = WMMA_F32(ar3, br0, acc[3][0]);
    acc[3][1] = WMMA_F32(ar3, br1, acc[3][1]);
    acc[0][0] = WMMA_F32(ai0, bi0, acc[0][0]);
    acc[0][1] = WMMA_F32(ai0, bi1, acc[0][1]);
    acc[1][0] = WMMA_F32(ai1, bi0, acc[1][0]);
    acc[1][1] = WMMA_F32(ai1, bi1, acc[1][1]);
    acc[2][0] = WMMA_F32(ai2, bi0, acc[2][0]);
    acc[2][1] = WMMA_F32(ai2, bi1, acc[2][1]);
    acc[3][0] = WMMA_F32(ai3, bi0, acc[3][0]);
    acc[3][1] = WMMA_F32(ai3, bi1, acc[3][1]);

    const int hi = lane >> 4;
#pragma unroll
    for (int j = 0; j < 4; ++j) {
        const int rbase = Mquad + j * 16 + 8 * hi;
#pragma unroll
        for (int n = 0; n < 2; ++n) {
            const int col = Nbase + n * 16 + mr;
            const float d = D[col];
#pragma unroll
            for (int i = 0; i < 8; ++i) {
                const size_t o = (size_t)(rbase + i) * H_DIM + col;
                out[o] = 2.0f * acc[j][n][i] + u[o] * d;
            }
        }
    }
}

// ---------------------------------------------------------------------------
// Host launcher
// ---------------------------------------------------------------------------
extern "C" void kernel_launch(void* const* d_in, const int* in_sizes, int n_in,
                              void* d_out, int out_size, void* d_ws, size_t ws_size,
                              hipStream_t stream)
{
    const float* hidden  = (const float*)d_in[0];
    const float* u       = (const float*)d_in[1];
    const float* Lam_re  = (const float*)d_in[2];
    const float* Lam_im  = (const float*)d_in[3];
    const float* B       = (const float*)d_in[4];
    const float* C       = (const float*)d_in[5];
    const float* D       = (const float*)d_in[6];
    const float* logstep = (const float*)d_in[7];
    const int*   resets  = (const int*)d_in[8];
    float* out = (float*)d_out;

    // workspace layout (floats)
    float* ws = (float*)d_ws;
    size_t o = 0;
    float* Bbar_re = ws + o; o += (size_t)P_DIM * H_DIM;
    float* Bbar_im = ws + o; o += (size_t)P_DIM * H_DIM;
    float* LamB_re = ws + o; o += P_DIM;
    float* LamB_im = ws + o; o += P_DIM;
    float* Cw_re   = ws + o; o += (size_t)H_DIM * P_DIM;
    float* Cw_imn  = ws + o; o += (size_t)H_DIM * P_DIM;
    float* BuRe    = ws + o; o += (size_t)L_SEQ * P_DIM;
    float* BuIm    = ws + o; o += (size_t)L_SEQ * P_DIM;
    float* Xre     = ws + o; o += (size_t)L_SEQ * P_DIM;
    float* Xim     = ws + o; o += (size_t)L_SEQ * P_DIM;
    float* AggAre  = ws + o; o += (size_t)NCHUNK * P_DIM;
    float* AggAim  = ws + o; o += (size_t)NCHUNK * P_DIM;
    float* AggBre  = ws + o; o += (size_t)NCHUNK * P_DIM;
    float* AggBim  = ws + o; o += (size_t)NCHUNK * P_DIM;
    float* AggC    = ws + o; o += (size_t)NCHUNK * P_DIM;
    float* XinRe   = ws + o; o += (size_t)NCHUNK * P_DIM;
    float* XinIm   = ws + o; o += (size_t)NCHUNK * P_DIM;

    // 1) prep
    s5_prep_b<<<(P_DIM * H_DIM + 255) / 256, 256, 0, stream>>>(
        Lam_re, Lam_im, B, logstep, Bbar_re, Bbar_im, LamB_re, LamB_im);
    s5_prep_c<<<(H_DIM * P_DIM + 255) / 256, 256, 0, stream>>>(C, Cw_re, Cw_imn);

    // 2) GEMM1: Bu (re, im); 4 M-tiles per wave, 8 waves per block
    {
        dim3 grid(P_DIM / 16, L_SEQ / (64 * 8));
        s5_gemm_bu<<<grid, 256, 0, stream>>>(u, Bbar_re, Bbar_im, BuRe, BuIm);
    }

    // 3) two-level chunked reset scan
    s5_scan_agg<<<NCHUNK, P_DIM, 0, stream>>>(BuRe, BuIm, LamB_re, LamB_im, resets,
                                              AggAre, AggAim, AggBre, AggBim, AggC);
    s5_scan_spine<<<1, P_DIM, 0, stream>>>(hidden, AggAre, AggAim, AggBre, AggBim,
                                           AggC, XinRe, XinIm);
    s5_scan_apply<<<NCHUNK, P_DIM, 0, stream>>>(BuRe, BuIm, LamB_re, LamB_im, resets,
                                                XinRe, XinIm, Xre, Xim);

    // 4) GEMM2 + epilogue; 4 M-tiles x 2 N-tiles per wave
    {
        dim3 grid(H_DIM / (16 * 2), L_SEQ / (64 * 8));
        s5_gemm_out<<<grid, 256, 0, stream>>>(Xre, Xim, Cw_re, Cw_imn, u, D, out);
    }
}